// PolarEncoder_22686017257974
// MI455X (gfx1250) — compile-verified
//
#include <hip/hip_runtime.h>

#define B_  1024
#define N_  8192
#define K_  4096
#define NB_ (B_ * N_)   // 8388608

typedef __attribute__((ext_vector_type(4))) int   v4i;
typedef __attribute__((ext_vector_type(2))) float v2f;
typedef __attribute__((ext_vector_type(4))) float v4f;
typedef __attribute__((address_space(1))) v4i* gv4i_p;   // global int4*
typedef __attribute__((address_space(3))) v4i* lv4i_p;   // LDS int4*

// ---------------------------------------------------------------------------
// Kernel 1/2: build inverse map of info_set in scratch: map[n] = k or -1
// ---------------------------------------------------------------------------
__global__ void k_map_init(int* __restrict__ map) {
    int n = blockIdx.x * 256 + threadIdx.x;
    if (n < N_) map[n] = -1;
}

__global__ void k_map_scatter(const int* __restrict__ info_set, int* __restrict__ map) {
    int k = blockIdx.x * 256 + threadIdx.x;
    if (k < K_) map[info_set[k]] = k;
}

// ---------------------------------------------------------------------------
// Kernel 3: elementwise outputs f,u,p,r (all coalesced, NT stores) +
// ballot-packed u2d bits.
// out layout (floats): x[0, NB) f[NB, 2NB) u[2NB, 3NB) p[3NB, 5NB) r[5NB, 7NB)
// ---------------------------------------------------------------------------
__global__ void __launch_bounds__(256)
k_elem(const float* __restrict__ info_bits,
       const int*   __restrict__ u_random,
       const int*   __restrict__ map,
       float*       __restrict__ out,
       unsigned int* __restrict__ pw) {
    int idx = blockIdx.x * 256 + threadIdx.x;   // 0 .. NB-1
    int b = idx >> 13;                          // / 8192
    int n = idx & (N_ - 1);

    int ur = u_random[idx];                     // 0/1
    int m  = map[n];                            // L2-resident, reused by all b
    int u2, f2;
    if (m >= 0) { u2 = (int)info_bits[b * K_ + m]; f2 = 2; }
    else        { u2 = ur;                        f2 = ur; }

    // streaming outputs: write-once, never re-read -> non-temporal
    __builtin_nontemporal_store((float)f2, out + NB_     + idx);   // f
    __builtin_nontemporal_store((float)u2, out + 2 * NB_ + idx);   // u

    v2f pv = { 0.5f, 0.5f };
    __builtin_nontemporal_store(pv, (v2f*)(out + 3 * NB_) + idx);  // p

    v2f rv = { (float)(1 - ur), (float)ur };
    __builtin_nontemporal_store(rv, (v2f*)(out + 5 * NB_) + idx);  // r

    // wave32-native bit packing: lane l of this wave holds element (base+l)
    unsigned int wbits = __builtin_amdgcn_ballot_w32((u2 & 1) != 0);
    if ((threadIdx.x & 31) == 0) pw[idx >> 5] = wbits;  // RT: consumed next kernel
}

// ---------------------------------------------------------------------------
// Kernel 4: per-row polar transform on packed bits + bit-reversed coalesced
// write of x. One block (256 threads, 8 waves) per batch row; thread t owns
// the 32 bits of indices [t*32, t*32+32).
// ---------------------------------------------------------------------------
__global__ void __launch_bounds__(256)
k_butterfly(const unsigned int* __restrict__ pw, float* __restrict__ out_x) {
    __shared__ unsigned int  s_w[256];
    __shared__ unsigned char s_x[8192];

    const int t = threadIdx.x;
    const int b = blockIdx.x;
    const unsigned int* row = pw + b * 256;

    // ---- stage packed words into LDS (CDNA5 async path) ----
#if __has_builtin(__builtin_amdgcn_global_load_async_to_lds_b128)
    if (t < 64) {
        __builtin_amdgcn_global_load_async_to_lds_b128(
            (gv4i_p)(row + t * 4),
            (lv4i_p)(&s_w[t * 4]),
            0, 0);
    }
    asm volatile("s_wait_asynccnt 0" ::: "memory");
    __syncthreads();
#else
    if (t < 64) ((uint4*)s_w)[t] = ((const uint4*)row)[t];
    __syncthreads();
#endif
    unsigned int w = s_w[t];

    // ---- superset-XOR transform, 13 commuting stages ----
    // bits 0..4: intra-word (w bit i = element t*32+i)
    w ^= (w >> 1)  & 0x55555555u;
    w ^= (w >> 2)  & 0x33333333u;
    w ^= (w >> 4)  & 0x0F0F0F0Fu;
    w ^= (w >> 8)  & 0x00FF00FFu;
    w ^= (w >> 16) & 0x0000FFFFu;

    // bits 5..9: cross-lane within the wave32 (lane bits of t)
#pragma unroll
    for (int k = 0; k < 5; ++k) {
        unsigned int p = (unsigned int)__shfl_xor((int)w, 1 << k, 32);
        if (((t >> k) & 1) == 0) w ^= p;
    }

    // bits 10..12: across the 8 waves via LDS
#pragma unroll
    for (int k = 5; k < 8; ++k) {
        __syncthreads();
        s_w[t] = w;
        __syncthreads();
        unsigned int p = s_w[t ^ (1 << k)];
        if (((t >> k) & 1) == 0) w ^= p;
    }

    // ---- bit-reversal x[o] = w_transform[rev13(o)] via LDS byte scatter ----
    // j = t*32+i -> o = rev5(i)*256 + rev8(t): conflict-free byte stores
    const int rev8t = (int)(__brev((unsigned)t) >> 24);
    __syncthreads();
#pragma unroll
    for (int i = 0; i < 32; ++i) {
        int rev5i = (int)(__brev((unsigned)i) >> 27);
        s_x[(rev5i << 8) | rev8t] = (unsigned char)((w >> i) & 1u);
    }
    __syncthreads();

    // ---- coalesced non-temporal float4 stores of x ----
    const unsigned int* sx32 = (const unsigned int*)s_x;
    v4f* ox = (v4f*)(out_x + b * N_);
#pragma unroll
    for (int j = 0; j < 8; ++j) {
        unsigned int v = sx32[j * 256 + t];
        v4f xv = { (float)( v        & 0xffu), (float)((v >> 8)  & 0xffu),
                   (float)((v >> 16) & 0xffu), (float)((v >> 24) & 0xffu) };
        __builtin_nontemporal_store(xv, ox + j * 256 + t);
    }
}

// ---------------------------------------------------------------------------
extern "C" void kernel_launch(void* const* d_in, const int* in_sizes, int n_in,
                              void* d_out, int out_size, void* d_ws, size_t ws_size,
                              hipStream_t stream) {
    const float* info_bits = (const float*)d_in[0];   // (B,K) f32
    const int*   u_random  = (const int*)  d_in[1];   // (B,N) i32
    const int*   info_set  = (const int*)  d_in[2];   // (K,)  i32
    float* out = (float*)d_out;                       // 7*B*N f32

    int*          map = (int*)d_ws;                   // N ints
    unsigned int* pw  = (unsigned int*)d_ws + N_;     // B*256 words

    k_map_init   <<<N_ / 256,  256, 0, stream>>>(map);
    k_map_scatter<<<K_ / 256,  256, 0, stream>>>(info_set, map);
    k_elem       <<<NB_ / 256, 256, 0, stream>>>(info_bits, u_random, map, out, pw);
    k_butterfly  <<<B_,        256, 0, stream>>>(pw, out);
}